// NGCF_57844619542655
// MI455X (gfx1250) — compile-verified
//
#include <hip/hip_runtime.h>

typedef float v2f __attribute__((ext_vector_type(2)));
typedef float v8f __attribute__((ext_vector_type(8)));

#define GROWS 128   // rows per GEMM block
#define RSTR  66    // padded LDS row stride (floats): stride-64 would 16-way bank conflict

__device__ __forceinline__ const float* ego_row(const float* egoU, const float* egoI,
                                                int nU, int r) {
  return (r < nU) ? (egoU + (long long)r * 64)
                  : (egoI + (long long)(r - nU) * 64);
}

// ---------------------------------------------------------------- zero side
__global__ __launch_bounds__(256) void zero_kernel(float4* __restrict__ p, long long n4) {
  long long i = (long long)blockIdx.x * 256 + threadIdx.x;
  if (i < n4) p[i] = make_float4(0.f, 0.f, 0.f, 0.f);
}

// ------------------------------------------------- SpMM scatter (L2 atomics)
// 32 lanes per edge, float2 per lane -> 2 native f32 atomic adds per lane.
__global__ __launch_bounds__(256) void spmm_scatter(
    const int* __restrict__ row, const int* __restrict__ col,
    const float* __restrict__ val,
    const float* __restrict__ egoU, const float* __restrict__ egoI,
    float* __restrict__ side, long long nnz, int nU) {
  long long g = (long long)blockIdx.x * 256 + threadIdx.x;
  long long e = g >> 5;
  int t = (int)(g & 31);
  if (e >= nnz) return;
  int r = row[e];
  int c = col[e];
  float v = val[e];
  const float2* src = (const float2*)ego_row(egoU, egoI, nU, c);
  float2 x = src[t];
  float* dst = side + (long long)r * 64 + t * 2;
  __hip_atomic_fetch_add(dst,     v * x.x, __ATOMIC_RELAXED, __HIP_MEMORY_SCOPE_AGENT);
  __hip_atomic_fetch_add(dst + 1, v * x.y, __ATOMIC_RELAXED, __HIP_MEMORY_SCOPE_AGENT);
}

// --------------------------------------------- fused dual-GEMM via WMMA f32
// outEgo = side @ W1^T + (ego .* side) @ W2^T + (b1 + b2)
// out[m][n] = sum_k S[m][k]*W1[n][k] + sum_k (E[m][k]*S[m][k])*W2[n][k]
__global__ __launch_bounds__(256) void gemm_fused(
    const float* __restrict__ side,
    const float* __restrict__ egoU, const float* __restrict__ egoI,
    const float* __restrict__ W1, const float* __restrict__ W2,
    const float* __restrict__ b1, const float* __restrict__ b2,
    float* __restrict__ outEgo, int Ntot, int nU) {
  __shared__ float sS[GROWS * RSTR];
  __shared__ float sE[GROWS * RSTR];
  __shared__ float sW1[64 * RSTR];
  __shared__ float sW2[64 * RSTR];
  __shared__ float sB[64];

  const int tid = threadIdx.x;
  const long long rowBase = (long long)blockIdx.x * GROWS;

  // Stage weights: 4096 floats each, float4 global loads, scalar LDS stores
  // (padded rows are not 16B aligned).
  for (int i = 0; i < 4; ++i) {
    int idx = tid + i * 256;                  // float4 chunk id, 0..1023
    int r = idx >> 4, c4 = (idx & 15) * 4;
    float4 w1 = ((const float4*)W1)[idx];
    float4 w2 = ((const float4*)W2)[idx];
    float* d1 = &sW1[r * RSTR + c4];
    d1[0] = w1.x; d1[1] = w1.y; d1[2] = w1.z; d1[3] = w1.w;
    float* d2 = &sW2[r * RSTR + c4];
    d2[0] = w2.x; d2[1] = w2.y; d2[2] = w2.z; d2[3] = w2.w;
  }
  if (tid < 64) sB[tid] = b1[tid] + b2[tid];

  // Stage 128 rows of side and ego (zero-padded past Ntot).
  for (int i = 0; i < 8; ++i) {
    int idx = tid + i * 256;                  // float4 chunk id, 0..2047
    int r = idx >> 4, cc = idx & 15;
    long long gr = rowBase + r;
    float4 sv = make_float4(0.f, 0.f, 0.f, 0.f);
    float4 ev = sv;
    if (gr < Ntot) {
      sv = ((const float4*)(side + gr * 64))[cc];
      const float* ep = ego_row(egoU, egoI, nU, (int)gr);
      ev = ((const float4*)ep)[cc];
    }
    float* ds_ = &sS[r * RSTR + cc * 4];
    ds_[0] = sv.x; ds_[1] = sv.y; ds_[2] = sv.z; ds_[3] = sv.w;
    float* de_ = &sE[r * RSTR + cc * 4];
    de_[0] = ev.x; de_[1] = ev.y; de_[2] = ev.z; de_[3] = ev.w;
  }
  __syncthreads();

  const int lane = tid & 31;
  const int wid  = tid >> 5;                  // wave id: 16-row strip
  const int mrow = lane & 15;                 // A/B/C lane row/col index
  const int kh   = lane >> 4;                 // K half-select per ISA layout

  v8f acc[4];
  acc[0] = v8f{}; acc[1] = v8f{}; acc[2] = v8f{}; acc[3] = v8f{};

  const float* aBase = &sS[(wid * 16 + mrow) * RSTR];
  const float* eBase = &sE[(wid * 16 + mrow) * RSTR];

#pragma unroll
  for (int kk = 0; kk < 16; ++kk) {           // K = 64 in steps of 4
    const int kb = kk * 4 + kh * 2;           // lanes 0-15: K={0,1}; 16-31: K={2,3}
    v2f aS, aP;
    aS[0] = aBase[kb]; aS[1] = aBase[kb + 1];
    aP[0] = aS[0] * eBase[kb];
    aP[1] = aS[1] * eBase[kb + 1];
#pragma unroll
    for (int j = 0; j < 4; ++j) {             // N tiles: 0,16,32,48
      const int n = j * 16 + mrow;
      v2f bB1, bB2;
      bB1[0] = sW1[n * RSTR + kb]; bB1[1] = sW1[n * RSTR + kb + 1];
      bB2[0] = sW2[n * RSTR + kb]; bB2[1] = sW2[n * RSTR + kb + 1];
      acc[j] = __builtin_amdgcn_wmma_f32_16x16x4_f32(
          false, aS, false, bB1, (short)0, acc[j], false, false);
      acc[j] = __builtin_amdgcn_wmma_f32_16x16x4_f32(
          false, aP, false, bB2, (short)0, acc[j], false, false);
    }
  }

  // Bias + store. C layout: VGPR v, lanes 0-15 -> M=v ; lanes 16-31 -> M=v+8.
#pragma unroll
  for (int j = 0; j < 4; ++j) {
    const int n = j * 16 + mrow;
    const float bv = sB[n];
    const long long rb = rowBase + wid * 16 + kh * 8;
#pragma unroll
    for (int v = 0; v < 8; ++v) {
      long long r = rb + v;
      if (r < Ntot) outEgo[r * 64 + n] = acc[j][v] + bv;
    }
  }
}

// ------------------------------------- fused gather + row-normalize (output)
// One wave per (sample, layer-slab). Slab 0 = raw embedding (unnormalized,
// matching all_embs[0]); slabs 1..3 normalized by row L2 norm.
__global__ __launch_bounds__(256) void gather_norm(
    const int* __restrict__ users, const int* __restrict__ pos,
    const int* __restrict__ neg,
    const float* __restrict__ embU, const float* __restrict__ embI,
    const float* __restrict__ e1, const float* __restrict__ e2,
    const float* __restrict__ e3,
    float* __restrict__ out, int nU, int B) {
  int gw = (int)(((long long)blockIdx.x * 256 + threadIdx.x) >> 5);
  int lane = threadIdx.x & 31;
  int slab = gw & 3;
  int samp = gw >> 2;
  if (samp >= 3 * B) return;
  int grp = samp / B;
  int i   = samp - grp * B;
  int idx = (grp == 0) ? users[i] : (nU + ((grp == 1) ? pos[i] : neg[i]));

  const float* src;
  if (slab == 0) {
    src = ego_row(embU, embI, nU, idx);
  } else {
    const float* e = (slab == 1) ? e1 : (slab == 2) ? e2 : e3;
    src = e + (long long)idx * 64;
  }
  float2 x = ((const float2*)src)[lane];

  float scale = 1.f;
  if (slab != 0) {
    float ss = x.x * x.x + x.y * x.y;
#pragma unroll
    for (int o = 16; o; o >>= 1) ss += __shfl_xor(ss, o, 32);
    scale = 1.f / fmaxf(sqrtf(ss), 1e-12f);
  }
  float* dst = out + (long long)samp * 256 + slab * 64 + lane * 2;
  dst[0] = x.x * scale;
  dst[1] = x.y * scale;
}

// ---------------------------------------------------------------------------
extern "C" void kernel_launch(void* const* d_in, const int* in_sizes, int n_in,
                              void* d_out, int out_size, void* d_ws, size_t ws_size,
                              hipStream_t stream) {
  const int*   users = (const int*)d_in[0];
  const int*   pos   = (const int*)d_in[1];
  const int*   neg   = (const int*)d_in[2];
  const int*   arow  = (const int*)d_in[3];
  const int*   acol  = (const int*)d_in[4];
  const float* aval  = (const float*)d_in[5];
  const float* uemb  = (const float*)d_in[6];
  const float* iemb  = (const float*)d_in[7];
  const float* gcnW  = (const float*)d_in[8];
  const float* gcnb  = (const float*)d_in[9];
  const float* biW   = (const float*)d_in[10];
  const float* bib   = (const float*)d_in[11];

  const int B        = in_sizes[0];
  const long long nnz = in_sizes[3];
  const int nU       = in_sizes[6] / 64;
  const int nI       = in_sizes[7] / 64;
  const int N        = nU + nI;
  const int L        = in_sizes[8] / (64 * 64);   // = 3

  // Workspace: 3 un-normalized ego slabs (layers 1..3) + side. 4 * N*64*4B.
  float* ws = (float*)d_ws;
  const long long slab = (long long)N * 64;
  float* ego1 = ws;
  float* ego2 = ws + slab;
  float* ego3 = ws + 2 * slab;
  float* sideb = ws + 3 * slab;
  float* egoL[4] = {nullptr, ego1, ego2, ego3};

  const int zb = (int)((N * 16LL + 255) / 256);          // float4 zero blocks
  const int sb = (int)((nnz * 32 + 255) / 256);          // scatter blocks
  const int mb = (N + GROWS - 1) / GROWS;                // gemm blocks

  for (int k = 0; k < L; ++k) {
    const float* pU = (k == 0) ? uemb : egoL[k];
    const float* pI = (k == 0) ? iemb : egoL[k] + (long long)nU * 64;
    zero_kernel<<<zb, 256, 0, stream>>>((float4*)sideb, N * 16LL);
    spmm_scatter<<<sb, 256, 0, stream>>>(arow, acol, aval, pU, pI, sideb, nnz, nU);
    gemm_fused<<<mb, 256, 0, stream>>>(sideb, pU, pI,
                                       gcnW + (long long)k * 4096,
                                       biW  + (long long)k * 4096,
                                       gcnb + (long long)k * 64,
                                       bib  + (long long)k * 64,
                                       egoL[k + 1], N, nU);
  }

  const int gwaves  = 3 * B * 4;
  const int gblocks = (gwaves * 32 + 255) / 256;
  gather_norm<<<gblocks, 256, 0, stream>>>(users, pos, neg, uemb, iemb,
                                           ego1, ego2, ego3, (float*)d_out, nU, B);
}